// HashTable_35338990911985
// MI455X (gfx1250) — compile-verified
//
#include <hip/hip_runtime.h>

// ---------------------------------------------------------------------------
// Multi-resolution hash-grid encoding (Instant-NGP style, distance weights).
//   coords      : (4096*192, 3)  f32
//   hash_tables : (8, 2^22, 2)   f32
//   out         : (4096*192, 16) f32   (8 levels x 2 features)
//
// Gather-bound: 403 MB of random 8B gathers, ~315 MB HBM-side -> ~13.5us at
// 23.3 TB/s. WMMA does not apply (per-lane gathered "B"). MI455X levers:
// wave32 occupancy, deep LOADcnt gather clauses, integer AND for the 2^22
// modulus (bit-exact vs the fp32 reference path), TRANS sqrt overlapped with
// gather latency, NT stores to keep the 192MB L2 holding the hot hash table.
// ---------------------------------------------------------------------------

#define HG_LEVELS     8
#define HG_TABLE_SIZE (1 << 22)   // 4194304

typedef float v2f __attribute__((ext_vector_type(2)));
typedef float v4f __attribute__((ext_vector_type(4)));

__global__ __launch_bounds__(256)
void hashgrid_encode_kernel(const float* __restrict__ coords,
                            const float* __restrict__ tables,
                            float* __restrict__ out,
                            int npts)
{
    const int p = blockIdx.x * blockDim.x + threadIdx.x;
    if (p >= npts) return;

    // Coalesced coord load (compiler merges into global_load_b96).
    const float cx = coords[3 * p + 0];
    const float cy = coords[3 * p + 1];
    const float cz = coords[3 * p + 2];

    const float P0 = 73856093.0f, P1 = 19349663.0f, P2 = 83492791.0f;
    const float OFF = 0.05f;          // VOXEL_SIZE (corner offsets are NOT per-level scaled)
    const float INV_DIAG = 1.0f / (1.7320508075688772f * 0.05f);

    // One IEEE divide per dim; q_l = q * 2^l is bit-exact vs c / vs_l
    // (rounding commutes with exact power-of-2 scaling).
    const float qx = cx / 0.05f;
    const float qy = cy / 0.05f;
    const float qz = cz / 0.05f;

    float feats[2 * HG_LEVELS];

#pragma unroll
    for (int l = 0; l < HG_LEVELS; ++l) {
        const float sl   = (float)(1 << l);
        const float vs_l = 0.05f / sl;                 // constant-folded, exact

        // base = float(int32(floor(c/vs))) * vs; int round-trip is identity.
        const float bx = floorf(qx * sl) * vs_l;
        const float by = floorf(qy * sl) * vs_l;
        const float bz = floorf(qz * sl) * vs_l;

        const float vx[2] = { bx, bx + OFF };
        const float vy[2] = { by, by + OFF };
        const float vz[2] = { bz, bz + OFF };

        // Hash partial products / sums — pinned rounding (no contraction):
        // h = (vx*P0 + vy*P1) + vz*P2 left-associated; hxy combos shared
        // across corners are bit-identical to per-corner evaluation.
        const float hx[2] = { __fmul_rn(vx[0], P0), __fmul_rn(vx[1], P0) };
        const float hy[2] = { __fmul_rn(vy[0], P1), __fmul_rn(vy[1], P1) };
        const float hz[2] = { __fmul_rn(vz[0], P2), __fmul_rn(vz[1], P2) };

        float hxy[2][2];
#pragma unroll
        for (int a = 0; a < 2; ++a)
#pragma unroll
            for (int b = 0; b < 2; ++b)
                hxy[a][b] = __fadd_rn(hx[a], hy[b]);

        // --- 1) indices first, so gathers can issue ASAP -----------------
        // idx = int32(clip(mod(h, 2^22), 0, 2^22-1))
        //     = trunc(h) & 0x3FFFFF   for 0 <= h < 2^28  (bit-exact:
        //       h = k*2^22 + r exactly; trunc removes only frac(r) < 1,
        //       which cannot cross a 2^22 boundary, and h >= 0).
        int idx[8];
#pragma unroll
        for (int c = 0; c < 8; ++c) {
            // OFFSETS order: (0,0,0),(1,0,0),(1,1,0),(0,1,0),
            //                (0,0,1),(1,0,1),(1,1,1),(0,1,1)
            const int ox = ((c + 1) >> 1) & 1;
            const int oy = (c >> 1) & 1;
            const int oz = (c >> 2) & 1;

            const float h = __fadd_rn(hxy[ox][oy], hz[oz]);
            idx[c] = ((int)h) & (HG_TABLE_SIZE - 1);
        }

        // --- 2) issue the 8-gather clause (global_load_b64 x8) -----------
        const v2f* __restrict__ tl = (const v2f*)tables + ((size_t)l << 22);
        v2f f[8];
#pragma unroll
        for (int c = 0; c < 8; ++c) f[c] = tl[idx[c]];

        // --- 3) distances/sqrt (TRANS) overlap the gather latency --------
        const float dx[2] = { cx - vx[0], cx - vx[1] };
        const float dy[2] = { cy - vy[0], cy - vy[1] };
        const float dz[2] = { cz - vz[0], cz - vz[1] };
        const float sx[2] = { dx[0] * dx[0], dx[1] * dx[1] };
        float tyz[2][2];
#pragma unroll
        for (int a = 0; a < 2; ++a)
#pragma unroll
            for (int b = 0; b < 2; ++b)
                tyz[a][b] = fmaf(dy[a], dy[a], dz[b] * dz[b]);

        float w[8];
#pragma unroll
        for (int c = 0; c < 8; ++c) {
            const int ox = ((c + 1) >> 1) & 1;
            const int oy = (c >> 1) & 1;
            const int oz = (c >> 2) & 1;
            w[c] = sqrtf(sx[ox] + tyz[oy][oz]);        // un-normalized distance
        }

        // --- 4) scalar dual-issue-friendly accumulate, deferred 1/DIAG ---
        float a0 = 0.0f, a1 = 0.0f;
#pragma unroll
        for (int c = 0; c < 8; ++c) {
            a0 = fmaf(f[c].x, w[c], a0);
            a1 = fmaf(f[c].y, w[c], a1);
        }
        feats[2 * l + 0] = a0 * INV_DIAG;
        feats[2 * l + 1] = a1 * INV_DIAG;
    }

    // 64B per point, four 16B non-temporal stores: stream 50MB of output
    // past L2 so the hash tables stay resident.
    v4f* __restrict__ o = (v4f*)(out + (size_t)p * 16);
#pragma unroll
    for (int i = 0; i < 4; ++i) {
        v4f v = { feats[4 * i + 0], feats[4 * i + 1],
                  feats[4 * i + 2], feats[4 * i + 3] };
        __builtin_nontemporal_store(v, o + i);
    }
}

extern "C" void kernel_launch(void* const* d_in, const int* in_sizes, int n_in,
                              void* d_out, int out_size, void* d_ws, size_t ws_size,
                              hipStream_t stream) {
    const float* coords = (const float*)d_in[0];   // (4096*192, 3) f32
    const float* tables = (const float*)d_in[1];   // (8, 2^22, 2)  f32
    float* out = (float*)d_out;                    // (4096*192, 16) f32

    const int npts = in_sizes[0] / 3;              // 786432
    const int block = 256;                         // 8 wave32s per workgroup
    const int grid  = (npts + block - 1) / block;  // 3072

    hashgrid_encode_kernel<<<grid, block, 0, stream>>>(coords, tables, out, npts);
}